// SA_Layer_30571577213450
// MI455X (gfx1250) — compile-verified
//
#include <hip/hip_runtime.h>
#include <math.h>

typedef __attribute__((ext_vector_type(2))) float v2f;
typedef __attribute__((ext_vector_type(8))) float v8f;

#define BATCH 2
#define BN    1024
#define MDIM  128
#define HDIM  64
#define QKVW  192
#define KNB   16

// ---- WMMA f32 16x16x4: D = A(16x4) * B(4x16) + C(16x16) -------------------
// A frag: lane l<16 holds A[row=l, k={0,1}], lane l>=16 holds A[row=l-16, k={2,3}]
// B frag: lane l<16 holds B[k={0,1}, col=l], lane l>=16 holds B[k={2,3}, col=l-16]
// C/D   : lane l holds rows (l>=16 ? 8..15 : 0..7) of column (l & 15)
static __device__ __forceinline__ v8f wmma4_f32(v2f a, v2f b, v8f c) {
#if __has_builtin(__builtin_amdgcn_wmma_f32_16x16x4_f32)
    return __builtin_amdgcn_wmma_f32_16x16x4_f32(false, a, false, b, (short)0, c,
                                                 false, false);
#else
    // scalar fallback (keeps compilation alive; histogram will show wmma==0)
    c[0] += a.x * b.x + a.y * b.y;
    return c;
#endif
}

// ---------------------------------------------------------------------------
// Kernel 1: qkv = x @ W_lin + b_lin      x:[2048,128], W:[128,192] -> [2048,192]
// One wave per 16x16 output tile. 4 waves / block.
// ---------------------------------------------------------------------------
__global__ __launch_bounds__(128) void qkv_gemm(const float* __restrict__ x,
                                                const float* __restrict__ Wl,
                                                const float* __restrict__ bl,
                                                float* __restrict__ qkv) {
    const int wave = threadIdx.x >> 5;
    const int lane = threadIdx.x & 31;
    const int NT   = QKVW / 16;               // 12 tiles in N
    const int tile = blockIdx.x * 4 + wave;   // 1536 tiles total
    const int mt = tile / NT, nt = tile % NT;
    if (mt >= (BATCH * BN) / 16) return;      // wave-uniform guard
    const int i0 = mt * 16, n0 = nt * 16;
    const int row = lane & 15;
    const int kb  = (lane >> 4) << 1;         // 0 or 2
    const int col = lane & 15;
    const int rb  = (lane >> 4) << 3;         // 0 or 8

    const float bias = bl[n0 + col];
    v8f acc;
#pragma unroll
    for (int r = 0; r < 8; ++r) acc[r] = bias;

#pragma unroll 4
    for (int k4 = 0; k4 < MDIM / 4; ++k4) {
        const int k = k4 * 4 + kb;
        v2f a, b;
        a.x = x[(i0 + row) * MDIM + k];
        a.y = x[(i0 + row) * MDIM + k + 1];
        b.x = Wl[(k    ) * QKVW + n0 + col];
        b.y = Wl[(k + 1) * QKVW + n0 + col];
        acc = wmma4_f32(a, b, acc);
    }
#pragma unroll
    for (int r = 0; r < 8; ++r)
        qkv[(i0 + rb + r) * QKVW + n0 + col] = acc[r];
}

// ---------------------------------------------------------------------------
// Kernel 2: pp = pos @ W_pos ; aa = pp + b_pos    pos:[2048,64], W:[64,64]
// ---------------------------------------------------------------------------
__global__ __launch_bounds__(128) void pos_gemm(const float* __restrict__ pos,
                                                const float* __restrict__ Wp,
                                                const float* __restrict__ bp,
                                                float* __restrict__ pp,
                                                float* __restrict__ aa) {
    const int wave = threadIdx.x >> 5;
    const int lane = threadIdx.x & 31;
    const int NT   = HDIM / 16;               // 4 tiles in N
    const int tile = blockIdx.x * 4 + wave;   // 512 tiles total
    const int mt = tile / NT, nt = tile % NT;
    if (mt >= (BATCH * BN) / 16) return;
    const int i0 = mt * 16, n0 = nt * 16;
    const int row = lane & 15;
    const int kb  = (lane >> 4) << 1;
    const int col = lane & 15;
    const int rb  = (lane >> 4) << 3;

    v8f acc;
#pragma unroll
    for (int r = 0; r < 8; ++r) acc[r] = 0.0f;

#pragma unroll
    for (int k4 = 0; k4 < HDIM / 4; ++k4) {
        const int k = k4 * 4 + kb;
        v2f a, b;
        a.x = pos[(i0 + row) * HDIM + k];
        a.y = pos[(i0 + row) * HDIM + k + 1];
        b.x = Wp[(k    ) * HDIM + n0 + col];
        b.y = Wp[(k + 1) * HDIM + n0 + col];
        acc = wmma4_f32(a, b, acc);
    }
    const float bias = bp[n0 + col];
#pragma unroll
    for (int r = 0; r < 8; ++r) {
        const float d = acc[r];
        pp[(i0 + rb + r) * HDIM + n0 + col] = d;
        aa[(i0 + rb + r) * HDIM + n0 + col] = d + bias;
    }
}

// ---------------------------------------------------------------------------
// Kernel 3: row norms  na[i]=||aa_i||^2, nc[i]=||pp_i||^2
// ---------------------------------------------------------------------------
__global__ __launch_bounds__(256) void row_norms(const float* __restrict__ pp,
                                                 const float* __restrict__ aa,
                                                 float* __restrict__ na,
                                                 float* __restrict__ nc) {
    const int i = blockIdx.x * 256 + threadIdx.x;
    if (i >= BATCH * BN) return;
    float sa = 0.f, sc = 0.f;
#pragma unroll 8
    for (int h = 0; h < HDIM; ++h) {
        const float av = aa[i * HDIM + h];
        const float cv = pp[i * HDIM + h];
        sa += av * av;
        sc += cv * cv;
    }
    na[i] = sa;
    nc[i] = sc;
}

// ---------------------------------------------------------------------------
// Kernel 4: fused distance-matrix (WMMA) -> LDS -> top-16 -> softmax attention
// One block = 4 waves handles 16 query rows of one batch.
//   Phase 1: wave w computes column tiles [w*256, w*256+256) of dist^2 via WMMA.
//   Phase 2: wave w selects top-16 + does attention for rows 4w..4w+3.
// ---------------------------------------------------------------------------
__global__ __launch_bounds__(128) void dist_topk_attn(
        const float* __restrict__ qkv, const float* __restrict__ pp,
        const float* __restrict__ aa,  const float* __restrict__ na,
        const float* __restrict__ nc,  float* __restrict__ out) {
    __shared__ float sdist[16 * BN];          // 64 KB (<< 320 KB/WGP)

    const int b    = blockIdx.x >> 6;         // 0..1
    const int i0   = (blockIdx.x & 63) << 4;  // 16-row block
    const int base = b * BN;
    const int wave = threadIdx.x >> 5;
    const int lane = threadIdx.x & 31;
    const int row  = lane & 15;
    const int kb   = (lane >> 4) << 1;
    const int col  = lane & 15;
    const int rb   = (lane >> 4) << 3;

    // Hoist the A-side (query rows) fragments for all 16 k-steps: 32 VGPRs.
    v2f afr[16];
#pragma unroll
    for (int k4 = 0; k4 < 16; ++k4) {
        const int k = k4 * 4 + kb;
        afr[k4].x = aa[(base + i0 + row) * HDIM + k];
        afr[k4].y = aa[(base + i0 + row) * HDIM + k + 1];
    }
    float nav[8];
#pragma unroll
    for (int r = 0; r < 8; ++r) nav[r] = na[base + i0 + rb + r];

    // ---- Phase 1: dist^2 tiles via WMMA ----
    for (int jt = wave * 16; jt < wave * 16 + 16; ++jt) {
        const int j0 = jt << 4;
        v8f acc;
#pragma unroll
        for (int r = 0; r < 8; ++r) acc[r] = 0.0f;
#pragma unroll
        for (int k4 = 0; k4 < 16; ++k4) {
            const int k = k4 * 4 + kb;
            v2f bf;
            bf.x = pp[(base + j0 + col) * HDIM + k];
            bf.y = pp[(base + j0 + col) * HDIM + k + 1];
            acc = wmma4_f32(afr[k4], bf, acc);
        }
        const float ncv = nc[base + j0 + col];
#pragma unroll
        for (int r = 0; r < 8; ++r) {
            // ||aa_i - pp_j||^2 = na_i + nc_j - 2 * <aa_i, pp_j>
            sdist[(rb + r) * BN + j0 + col] = nav[r] + ncv - 2.0f * acc[r];
        }
    }
    __syncthreads();

    // ---- Phase 2: per-row top-16 selection + vector attention ----
    for (int rr = 0; rr < 4; ++rr) {
        const int r = wave * 4 + rr;
        const int i = i0 + r;

        // Register-resident candidates: lane owns cols lane + 32*t.
        float v[32];
#pragma unroll
        for (int t = 0; t < 32; ++t) v[t] = sdist[r * BN + lane + 32 * t];

        int nbr[KNB];
        for (int ks = 0; ks < KNB; ++ks) {
            // local argmin (stable: keeps smallest t, i.e. smallest col, on tie)
            float bv = v[0];
            int   bt = 0;
#pragma unroll
            for (int t = 1; t < 32; ++t)
                if (v[t] < bv) { bv = v[t]; bt = t; }
            int bi = bt * 32 + lane;
            // wave32 min/argmin reduction, index tie-break -> matches top_k order
#pragma unroll
            for (int off = 16; off > 0; off >>= 1) {
                const float ov = __shfl_xor(bv, off, 32);
                const int   oi = __shfl_xor(bi, off, 32);
                if (ov < bv || (ov == bv && oi < bi)) { bv = ov; bi = oi; }
            }
            nbr[ks] = bi;                     // uniform across the wave
            const int selT = bi >> 5, selL = bi & 31;
            if (lane == selL) {
#pragma unroll
                for (int t = 0; t < 32; ++t)
                    if (t == selT) v[t] = 3.0e38f;
            }
        }

        // attention: each lane owns channels h = lane and h = lane + 32
#pragma unroll
        for (int hh = 0; hh < 2; ++hh) {
            const int h  = lane + hh * 32;
            const float qh  = qkv[(base + i) * QKVW + h];          // q
            const float aai = aa[(base + i) * HDIM + h];
            float s[KNB], vg[KNB];
            float m = -3.0e38f;
#pragma unroll
            for (int k = 0; k < KNB; ++k) {
                const int j  = nbr[k];
                const float kh = qkv[(base + j) * QKVW + 64 + h];  // k
                const float vh = qkv[(base + j) * QKVW + 128 + h]; // v
                const float pj = pp[(base + j) * HDIM + h];
                const float sv = qh - kh;                          // qk_rel
                s[k]  = sv;
                vg[k] = vh + (aai - pj);                           // v + rel_dis
                m = fmaxf(m, sv);
            }
            float den = 0.f, num = 0.f;
#pragma unroll
            for (int k = 0; k < KNB; ++k) {
                const float e = __expf(s[k] - m);
                den += e;
                num += e * vg[k];
            }
            out[(base + i) * HDIM + h] = num / den;
        }
    }
}

// ---------------------------------------------------------------------------
extern "C" void kernel_launch(void* const* d_in, const int* in_sizes, int n_in,
                              void* d_out, int out_size, void* d_ws, size_t ws_size,
                              hipStream_t stream) {
    const float* x   = (const float*)d_in[0];   // [2,1024,128]
    const float* pos = (const float*)d_in[1];   // [2,1024,64]
    const float* Wl  = (const float*)d_in[2];   // [128,192]
    const float* bl  = (const float*)d_in[3];   // [192]
    const float* Wp  = (const float*)d_in[4];   // [64,64]
    const float* bp  = (const float*)d_in[5];   // [64]
    float* out = (float*)d_out;                 // [2,1024,64]

    // workspace layout (~2.6 MB)
    float* qkv = (float*)d_ws;                            // 2048*192
    float* pp  = qkv + (size_t)BATCH * BN * QKVW;         // 2048*64
    float* aa  = pp  + (size_t)BATCH * BN * HDIM;         // 2048*64
    float* na  = aa  + (size_t)BATCH * BN * HDIM;         // 2048
    float* nc  = na  + (size_t)BATCH * BN;                // 2048

    // 1) qkv = x@W_lin + b_lin  (1536 WMMA tiles, 4 waves/block)
    qkv_gemm<<<384, 128, 0, stream>>>(x, Wl, bl, qkv);
    // 2) pp = pos@W_pos ; aa = pp + b_pos  (512 tiles)
    pos_gemm<<<128, 128, 0, stream>>>(pos, Wp, bp, pp, aa);
    // 3) row norms
    row_norms<<<(BATCH * BN + 255) / 256, 256, 0, stream>>>(pp, aa, na, nc);
    // 4) fused distance (WMMA) + top-16 + softmax attention
    dist_topk_attn<<<BATCH * (BN / 16), 128, 0, stream>>>(qkv, pp, aa, na, nc, out);
}